// DecoderAttentionRNN_60043642798485
// MI455X (gfx1250) — compile-verified
//
#include <hip/hip_runtime.h>
#include <hip/hip_bf16.h>
#include <math.h>

#define H   256
#define H2  512
#define V   8192
#define S   4096

typedef __attribute__((ext_vector_type(2))) float v2f;
typedef __attribute__((ext_vector_type(8))) float v8f;

// workspace layout (float offsets)
#define WS_G      0        // 256   : Wa@h + Wa_b + Ua_b
#define WS_EIJ    256      // 4096  : raw attention scores
#define WS_CI     4352     // 512   : context vector (atomic-accumulated)
#define WS_Z      4864     // 256
#define WS_R      5120     // 256
#define WS_LOGITS 5376     // 8192

// d_out layout: [0,8192) log_softmax ; [8192,8448) hidden_new ; [8448,12544) aij

// ---------------- K1: g = Wa@h + Wa_b + Ua_b ; zero Ci ----------------
__global__ void __launch_bounds__(256)
k_prep(const float* __restrict__ Wa_w, const float* __restrict__ Wa_b,
       const float* __restrict__ Ua_b, const float* __restrict__ hidden,
       float* __restrict__ ws) {
  int i = threadIdx.x;                       // 256 threads
  float acc = Wa_b[i] + Ua_b[i];
  for (int j = 0; j < H; ++j) acc += Wa_w[i * H + j] * hidden[j];
  ws[WS_G + i]       = acc;
  ws[WS_CI + i]      = 0.f;
  ws[WS_CI + H + i]  = 0.f;
}

// ---------------- K2: Eij[s] = Va . tanh(enc[s]@Ua^T + g) + Va_b ------
// WMMA f32 16x16x4. Block = 64 thr (2 waves); block handles 16 rows of enc;
// each wave owns 8 N-tiles of 16 columns (wave0: cols 0-127, wave1: 128-255).
// A layout (16x4 f32): lanes 0-15 -> M, VGPR.x/.y = K {0,1} (half 0) / {2,3} (half 1).
// B layout (4x16 f32): mirrored, lanes 0-15 -> N.
// C layout: VGPR v -> row v (lanes 0-15) / v+8 (lanes 16-31), lane&15 -> col.
__global__ void __launch_bounds__(64)
k_attn_scores(const float* __restrict__ enc,   // [S, 512]
              const float* __restrict__ Ua_w,  // [256, 512]
              const float* __restrict__ Va_w,  // [256]
              const float* __restrict__ Va_b,  // [1]
              float* __restrict__ ws) {
  const int r0   = blockIdx.x * 16;
  const int wave = threadIdx.x >> 5;
  const int lane = threadIdx.x & 31;
  const int half = lane >> 4;
  const int l16  = lane & 15;
  const int kk   = half * 2;
  const int colbase = wave * 128;

  const float* __restrict__ gvec = ws + WS_G;
  const float* __restrict__ arow = enc + (size_t)(r0 + l16) * H2;

  v8f acc[8];
#pragma unroll
  for (int t = 0; t < 8; ++t)
    acc[t] = (v8f){0.f, 0.f, 0.f, 0.f, 0.f, 0.f, 0.f, 0.f};

  for (int k0 = 0; k0 < H2; k0 += 4) {
    v2f a;
    a.x = arow[k0 + kk];
    a.y = arow[k0 + kk + 1];
#pragma unroll
    for (int t = 0; t < 8; ++t) {
      const float* __restrict__ brow = Ua_w + (size_t)(colbase + t * 16 + l16) * H2;
      v2f b;
      b.x = brow[k0 + kk];
      b.y = brow[k0 + kk + 1];
      acc[t] = __builtin_amdgcn_wmma_f32_16x16x4_f32(
          /*neg_a=*/false, a, /*neg_b=*/false, b,
          /*c_mod=*/(short)0, acc[t], /*reuse_a=*/false, /*reuse_b=*/false);
    }
  }

  // fused epilogue: rowacc[v] = sum_n tanh(C[m][n] + g[col]) * Va[col]
  float rowacc[8];
#pragma unroll
  for (int v = 0; v < 8; ++v) rowacc[v] = 0.f;
#pragma unroll
  for (int t = 0; t < 8; ++t) {
    const int colg = colbase + t * 16 + l16;
    const float gb = gvec[colg];
    const float va = Va_w[colg];
#pragma unroll
    for (int v = 0; v < 8; ++v)
      rowacc[v] += tanhf(acc[t][v] + gb) * va;
  }
  // reduce over the 16 lanes of each half (xor stays within the half)
#pragma unroll
  for (int v = 0; v < 8; ++v) {
    float x = rowacc[v];
    x += __shfl_xor(x, 1);
    x += __shfl_xor(x, 2);
    x += __shfl_xor(x, 4);
    x += __shfl_xor(x, 8);
    rowacc[v] = x;
  }
  __shared__ float red[2][16];
  if (l16 == 0) {
#pragma unroll
    for (int v = 0; v < 8; ++v) red[wave][half * 8 + v] = rowacc[v];
  }
  __syncthreads();
  if (threadIdx.x < 16)
    ws[WS_EIJ + r0 + threadIdx.x] =
        red[0][threadIdx.x] + red[1][threadIdx.x] + Va_b[0];
}

// ---------------- K3: aij = softmax(Eij) -> d_out[V+H ..] -------------
__global__ void __launch_bounds__(1024)
k_softmax(const float* __restrict__ ws, float* __restrict__ out) {
  __shared__ float red[1024];
  const int t = threadIdx.x;
  const float* __restrict__ E = ws + WS_EIJ;
  float v[4];
  float m = -INFINITY;
#pragma unroll
  for (int i = 0; i < 4; ++i) { v[i] = E[t + i * 1024]; m = fmaxf(m, v[i]); }
  red[t] = m; __syncthreads();
  for (int s = 512; s > 0; s >>= 1) {
    if (t < s) red[t] = fmaxf(red[t], red[t + s]);
    __syncthreads();
  }
  m = red[0]; __syncthreads();
  float sum = 0.f;
#pragma unroll
  for (int i = 0; i < 4; ++i) { v[i] = expf(v[i] - m); sum += v[i]; }
  red[t] = sum; __syncthreads();
  for (int s = 512; s > 0; s >>= 1) {
    if (t < s) red[t] += red[t + s];
    __syncthreads();
  }
  const float inv = 1.f / red[0];
#pragma unroll
  for (int i = 0; i < 4; ++i) out[V + H + t + i * 1024] = v[i] * inv;
}

// ---------------- K4: Ci = sum_s aij[s] * enc[s,:] --------------------
__global__ void __launch_bounds__(256)
k_context(const float* __restrict__ enc, const float* __restrict__ aij,
          float* __restrict__ ws) {
  const int j  = threadIdx.x;
  const int s0 = blockIdx.x * 128;          // grid = 32
  float acc0 = 0.f, acc1 = 0.f;
  for (int s = s0; s < s0 + 128; ++s) {
    const float a = aij[s];
    acc0 += a * enc[(size_t)s * H2 + j];
    acc1 += a * enc[(size_t)s * H2 + j + 256];
  }
  atomicAdd(&ws[WS_CI + j], acc0);
  atomicAdd(&ws[WS_CI + 256 + j], acc1);
}

// ---------------- K5: z, r gates (wave-per-output-row GEMV) -----------
__global__ void __launch_bounds__(256)
k_gates(const int* __restrict__ tok, const float* __restrict__ emb,
        const float* __restrict__ Uz_w, const float* __restrict__ Uz_b,
        const float* __restrict__ Wz_w, const float* __restrict__ Wz_b,
        const float* __restrict__ Cz_w, const float* __restrict__ Cz_b,
        const float* __restrict__ Ur_w, const float* __restrict__ Ur_b,
        const float* __restrict__ Wr_w, const float* __restrict__ Wr_b,
        const float* __restrict__ Cr_w, const float* __restrict__ Cr_b,
        const float* __restrict__ hidden, float* __restrict__ ws) {
  const int gw   = blockIdx.x * 8 + (threadIdx.x >> 5);  // 0..511 (grid = 64)
  const int lane = threadIdx.x & 31;
  const int isr  = gw >= H;
  const int row  = isr ? gw - H : gw;
  const float* U  = isr ? Ur_w : Uz_w;
  const float* W  = isr ? Wr_w : Wz_w;
  const float* C  = isr ? Cr_w : Cz_w;
  const float* Ub = isr ? Ur_b : Uz_b;
  const float* Wb = isr ? Wr_b : Wz_b;
  const float* Cb = isr ? Cr_b : Cz_b;
  const float* __restrict__ y  = emb + (size_t)tok[0] * V;
  const float* __restrict__ Ci = ws + WS_CI;

  float acc = 0.f;
  const float* Urow = U + (size_t)row * V;
  for (int j = lane; j < V; j += 32) acc += Urow[j] * y[j];
  const float* Wrow = W + row * H;
  for (int j = lane; j < H; j += 32) acc += Wrow[j] * hidden[j];
  const float* Crow = C + row * H2;
  for (int j = lane; j < H2; j += 32) acc += Crow[j] * Ci[j];
  for (int off = 16; off > 0; off >>= 1) acc += __shfl_xor(acc, off);
  if (lane == 0) {
    const float x = acc + Ub[row] + Wb[row] + Cb[row];
    ws[(isr ? WS_R : WS_Z) + row] = 1.f / (1.f + expf(-x));
  }
}

// ---------------- K6: candidate c, hidden_new -> d_out[V ..] ----------
__global__ void __launch_bounds__(256)
k_cand(const int* __restrict__ tok, const float* __restrict__ emb,
       const float* __restrict__ Uh_w, const float* __restrict__ Uh_b,
       const float* __restrict__ Wh_w, const float* __restrict__ Wh_b,
       const float* __restrict__ Ch_w, const float* __restrict__ Ch_b,
       const float* __restrict__ hidden, float* __restrict__ ws,
       float* __restrict__ out) {
  const int row  = blockIdx.x * 8 + (threadIdx.x >> 5);  // 0..255 (grid = 32)
  const int lane = threadIdx.x & 31;
  const float* __restrict__ y  = emb + (size_t)tok[0] * V;
  const float* __restrict__ Ci = ws + WS_CI;
  const float* __restrict__ r  = ws + WS_R;

  float acc = 0.f;
  const float* Urow = Uh_w + (size_t)row * V;
  for (int j = lane; j < V; j += 32) acc += Urow[j] * y[j];
  const float* Wrow = Wh_w + row * H;
  for (int j = lane; j < H; j += 32) acc += Wrow[j] * (r[j] * hidden[j]);
  const float* Crow = Ch_w + row * H2;
  for (int j = lane; j < H2; j += 32) acc += Crow[j] * Ci[j];
  for (int off = 16; off > 0; off >>= 1) acc += __shfl_xor(acc, off);
  if (lane == 0) {
    const float c = tanhf(acc + Uh_b[row] + Wh_b[row] + Ch_b[row]);
    const float z = ws[WS_Z + row];
    out[V + row] = (1.f - z) * c + z * hidden[row];
  }
}

// ---------------- K7: logits = V_w @ hidden_new + V_b -----------------
__global__ void __launch_bounds__(256)
k_logits(const float* __restrict__ V_w, const float* __restrict__ V_b,
         const float* __restrict__ out, float* __restrict__ ws) {
  const int row  = blockIdx.x * 8 + (threadIdx.x >> 5);  // 0..8191 (grid = 1024)
  const int lane = threadIdx.x & 31;
  const float* __restrict__ h = out + V;   // hidden_new
  float acc = 0.f;
  const float* Vrow = V_w + (size_t)row * H;
  for (int j = lane; j < H; j += 32) acc += Vrow[j] * h[j];
  for (int off = 16; off > 0; off >>= 1) acc += __shfl_xor(acc, off);
  if (lane == 0) ws[WS_LOGITS + row] = acc + V_b[row];
}

// ---------------- K8: out = log_softmax(logits) -----------------------
__global__ void __launch_bounds__(1024)
k_logsoftmax(const float* __restrict__ ws, float* __restrict__ out) {
  __shared__ float red[1024];
  const int t = threadIdx.x;
  const float* __restrict__ L = ws + WS_LOGITS;
  float v[8];
  float m = -INFINITY;
#pragma unroll
  for (int i = 0; i < 8; ++i) { v[i] = L[t + i * 1024]; m = fmaxf(m, v[i]); }
  red[t] = m; __syncthreads();
  for (int s = 512; s > 0; s >>= 1) {
    if (t < s) red[t] = fmaxf(red[t], red[t + s]);
    __syncthreads();
  }
  m = red[0]; __syncthreads();
  float sum = 0.f;
#pragma unroll
  for (int i = 0; i < 8; ++i) sum += expf(v[i] - m);
  red[t] = sum; __syncthreads();
  for (int s = 512; s > 0; s >>= 1) {
    if (t < s) red[t] += red[t + s];
    __syncthreads();
  }
  const float lse = m + logf(red[0]);
#pragma unroll
  for (int i = 0; i < 8; ++i) out[t + i * 1024] = v[i] - lse;
}

extern "C" void kernel_launch(void* const* d_in, const int* in_sizes, int n_in,
                              void* d_out, int out_size, void* d_ws, size_t ws_size,
                              hipStream_t stream) {
  const int*   tok    = (const int*)  d_in[0];
  const float* hidden = (const float*)d_in[1];
  const float* enc    = (const float*)d_in[2];
  const float* emb    = (const float*)d_in[3];
  const float* Uz_w = (const float*)d_in[4];  const float* Uz_b = (const float*)d_in[5];
  const float* Wz_w = (const float*)d_in[6];  const float* Wz_b = (const float*)d_in[7];
  const float* Cz_w = (const float*)d_in[8];  const float* Cz_b = (const float*)d_in[9];
  const float* Ur_w = (const float*)d_in[10]; const float* Ur_b = (const float*)d_in[11];
  const float* Wr_w = (const float*)d_in[12]; const float* Wr_b = (const float*)d_in[13];
  const float* Cr_w = (const float*)d_in[14]; const float* Cr_b = (const float*)d_in[15];
  const float* Uh_w = (const float*)d_in[16]; const float* Uh_b = (const float*)d_in[17];
  const float* Wh_w = (const float*)d_in[18]; const float* Wh_b = (const float*)d_in[19];
  const float* Ch_w = (const float*)d_in[20]; const float* Ch_b = (const float*)d_in[21];
  const float* Ua_w = (const float*)d_in[22]; const float* Ua_b = (const float*)d_in[23];
  const float* Wa_w = (const float*)d_in[24]; const float* Wa_b = (const float*)d_in[25];
  const float* Va_w = (const float*)d_in[26]; const float* Va_b = (const float*)d_in[27];
  const float* V_w  = (const float*)d_in[28]; const float* V_b  = (const float*)d_in[29];

  float* ws  = (float*)d_ws;
  float* out = (float*)d_out;

  k_prep<<<1, 256, 0, stream>>>(Wa_w, Wa_b, Ua_b, hidden, ws);
  k_attn_scores<<<S / 16, 64, 0, stream>>>(enc, Ua_w, Va_w, Va_b, ws);
  k_softmax<<<1, 1024, 0, stream>>>(ws, out);
  k_context<<<32, 256, 0, stream>>>(enc, out + V + H, ws);
  k_gates<<<64, 256, 0, stream>>>(tok, emb,
                                  Uz_w, Uz_b, Wz_w, Wz_b, Cz_w, Cz_b,
                                  Ur_w, Ur_b, Wr_w, Wr_b, Cr_w, Cr_b,
                                  hidden, ws);
  k_cand<<<32, 256, 0, stream>>>(tok, emb,
                                 Uh_w, Uh_b, Wh_w, Wh_b, Ch_w, Ch_b,
                                 hidden, ws, out);
  k_logits<<<1024, 256, 0, stream>>>(V_w, V_b, out, ws);
  k_logsoftmax<<<1, 1024, 0, stream>>>(ws, out);
}